// GCN_33483565039921
// MI455X (gfx1250) — compile-verified
//
#include <hip/hip_runtime.h>
#include <math.h>

typedef __attribute__((ext_vector_type(2))) float v2f;
typedef __attribute__((ext_vector_type(8))) float v8f;

#define WT_STRIDE 68  // padded LDS row stride (floats): banks (4*col+k)%64 -> conflict-free b64 loads

// ---------------- degree / normalization ----------------

__global__ __launch_bounds__(256) void k_deg(const int* __restrict__ dst,
                                             unsigned* __restrict__ deg, int E) {
  int e = blockIdx.x * 256 + threadIdx.x;
  if (e < E) atomicAdd(&deg[dst[e]], 1u);
}

__global__ __launch_bounds__(256) void k_dinv(const unsigned* __restrict__ deg,
                                              float* __restrict__ dinv, int n) {
  int i = blockIdx.x * 256 + threadIdx.x;
  if (i < n) dinv[i] = rsqrtf((float)(deg[i] + 1u));  // +1 = self-loop
}

// ---------------- GEMM1: hs = (x @ W[64x64]) * dinv[row]; dual-store to hs and agg ----------------

__global__ __launch_bounds__(256) void k_gemm1(const float* __restrict__ x,
                                               const float* __restrict__ W,
                                               const float* __restrict__ dinv,
                                               float* __restrict__ hs,
                                               float* __restrict__ agg, int n) {
  // W transposed in LDS: wt[col*WT_STRIDE + k] = W[k*64 + col]
  __shared__ float wt[64 * WT_STRIDE];
  for (int i = threadIdx.x; i < 64 * 64; i += 256) {
    int k = i >> 6, col = i & 63;
    wt[col * WT_STRIDE + k] = W[i];
  }
  __syncthreads();

  const int wave = threadIdx.x >> 5;
  const int lane = threadIdx.x & 31;
  const int m    = lane & 15;
  const int hi   = lane >> 4;      // lane half: 0 -> K 0/1, 1 -> K 2/3
  const int koff = hi * 2;
  const int row0 = (blockIdx.x * 8 + wave) * 16;
  if (row0 >= n) return;

  if (row0 + 16 <= n) {            // full 16-row tile: wave-uniform, EXEC all ones
    const int row = row0 + m;
    v8f c[4] = {};
#pragma unroll
    for (int k0 = 0; k0 < 64; k0 += 4) {
      float2 av = *(const float2*)(x + (size_t)row * 64 + k0 + koff);
      v2f a; a[0] = av.x; a[1] = av.y;
#pragma unroll
      for (int j = 0; j < 4; ++j) {
        // contiguous 8B LDS load -> adjacent aligned VGPR pair, no repacking
        v2f b = *(const v2f*)(&wt[(j * 16 + m) * WT_STRIDE + k0 + koff]);
        c[j] = __builtin_amdgcn_wmma_f32_16x16x4_f32(false, a, false, b,
                                                     (short)0, c[j], false, false);
      }
    }
    const int rbase = row0 + hi * 8;
#pragma unroll
    for (int r = 0; r < 8; ++r) {
      const float dv = dinv[rbase + r];
#pragma unroll
      for (int j = 0; j < 4; ++j) {
        const float v = c[j][r] * dv;
        const size_t o = (size_t)(rbase + r) * 64 + j * 16 + m;
        hs[o] = v;
        agg[o] = v;                 // self-loop term seeds the accumulator
      }
    }
  } else {                          // partial tile: scalar fallback (no WMMA)
    const int r = row0 + lane;
    if (lane < 16 && r < n) {
      const float dv = dinv[r];
      for (int cidx = 0; cidx < 64; ++cidx) {
        float acc = 0.f;
        for (int k = 0; k < 64; ++k) acc += x[(size_t)r * 64 + k] * wt[cidx * WT_STRIDE + k];
        const float v = acc * dv;
        hs[(size_t)r * 64 + cidx] = v;
        agg[(size_t)r * 64 + cidx] = v;
      }
    }
  }
}

// ---------------- GEMM2: hs = (h1 @ W[64x16]) * dinv[row] ----------------

__global__ __launch_bounds__(256) void k_gemm2(const float* __restrict__ h,
                                               const float* __restrict__ W,
                                               const float* __restrict__ dinv,
                                               float* __restrict__ hs,
                                               float* __restrict__ agg, int n) {
  __shared__ float wt[16 * WT_STRIDE];   // wt[col*WT_STRIDE + k] = W[k*16 + col]
  for (int i = threadIdx.x; i < 64 * 16; i += 256) {
    int k = i >> 4, col = i & 15;
    wt[col * WT_STRIDE + k] = W[i];
  }
  __syncthreads();

  const int wave = threadIdx.x >> 5;
  const int lane = threadIdx.x & 31;
  const int m    = lane & 15;
  const int hi   = lane >> 4;
  const int koff = hi * 2;
  const int row0 = (blockIdx.x * 8 + wave) * 16;
  if (row0 >= n) return;

  if (row0 + 16 <= n) {
    const int row = row0 + m;
    v8f c = {};
#pragma unroll
    for (int k0 = 0; k0 < 64; k0 += 4) {
      float2 av = *(const float2*)(h + (size_t)row * 64 + k0 + koff);
      v2f a; a[0] = av.x; a[1] = av.y;
      v2f b = *(const v2f*)(&wt[m * WT_STRIDE + k0 + koff]);
      c = __builtin_amdgcn_wmma_f32_16x16x4_f32(false, a, false, b,
                                                (short)0, c, false, false);
    }
    const int rbase = row0 + hi * 8;
#pragma unroll
    for (int r = 0; r < 8; ++r) {
      const float v = c[r] * dinv[rbase + r];
      const size_t o = (size_t)(rbase + r) * 16 + m;
      hs[o] = v;
      agg[o] = v;
    }
  } else {
    const int r = row0 + lane;
    if (lane < 16 && r < n) {
      const float dv = dinv[r];
      for (int cidx = 0; cidx < 16; ++cidx) {
        float acc = 0.f;
        for (int k = 0; k < 64; ++k) acc += h[(size_t)r * 64 + k] * wt[cidx * WT_STRIDE + k];
        const float v = acc * dv;
        hs[(size_t)r * 16 + cidx] = v;
        agg[(size_t)r * 16 + cidx] = v;
      }
    }
  }
}

// ---------------- edge scatter: agg[dst] += hs[src]  (feat = (1<<lshift)*4 floats) ----------------

__global__ __launch_bounds__(256) void k_scatter(const int* __restrict__ src,
                                                 const int* __restrict__ dst,
                                                 const float* __restrict__ hs,
                                                 float* __restrict__ agg,
                                                 int E, int lshift) {
  int t = blockIdx.x * 256 + threadIdx.x;
  int e = t >> lshift;
  if (e >= E) return;
  int l = t & ((1 << lshift) - 1);
  int s = src[e], d = dst[e];
  const int fw = (1 << lshift) * 4;
  float4 v = *(const float4*)(hs + (size_t)s * fw + l * 4);
  float* p = agg + (size_t)d * fw + l * 4;
  atomicAdd(p + 0, v.x);
  atomicAdd(p + 1, v.y);
  atomicAdd(p + 2, v.z);
  atomicAdd(p + 3, v.w);
}

// ---------------- bias + ReLU (layer 1 output, 64 feats) ----------------

__global__ __launch_bounds__(256) void k_bias_relu(const float* __restrict__ agg,
                                                   const float* __restrict__ dinv,
                                                   const float* __restrict__ b,
                                                   float* __restrict__ h, int n) {
  int i = blockIdx.x * 256 + threadIdx.x;
  if (i >= n * 64) return;
  int row = i >> 6, col = i & 63;
  float v = dinv[row] * agg[i] + b[col];
  h[i] = v > 0.f ? v : 0.f;
}

// ---------------- bias + log_softmax over 16 outputs ----------------

__global__ __launch_bounds__(256) void k_out(const float* __restrict__ agg,
                                             const float* __restrict__ dinv,
                                             const float* __restrict__ b,
                                             float* __restrict__ out, int n) {
  int i = blockIdx.x * 256 + threadIdx.x;
  if (i >= n) return;
  float dv = dinv[i];
  float z[16];
#pragma unroll
  for (int q = 0; q < 4; ++q) {
    float4 a  = *(const float4*)(agg + (size_t)i * 16 + q * 4);
    float4 bb = *(const float4*)(b + q * 4);
    z[q * 4 + 0] = dv * a.x + bb.x;
    z[q * 4 + 1] = dv * a.y + bb.y;
    z[q * 4 + 2] = dv * a.z + bb.z;
    z[q * 4 + 3] = dv * a.w + bb.w;
  }
  float mx = z[0];
#pragma unroll
  for (int k = 1; k < 16; ++k) mx = fmaxf(mx, z[k]);
  float s = 0.f;
#pragma unroll
  for (int k = 0; k < 16; ++k) s += expf(z[k] - mx);
  float lse = mx + logf(s);
#pragma unroll
  for (int q = 0; q < 4; ++q) {
    float4 o;
    o.x = z[q * 4 + 0] - lse;
    o.y = z[q * 4 + 1] - lse;
    o.z = z[q * 4 + 2] - lse;
    o.w = z[q * 4 + 3] - lse;
    *(float4*)(out + (size_t)i * 16 + q * 4) = o;
  }
}

// ---------------- launcher ----------------

extern "C" void kernel_launch(void* const* d_in, const int* in_sizes, int n_in,
                              void* d_out, int out_size, void* d_ws, size_t ws_size,
                              hipStream_t stream) {
  const float* x  = (const float*)d_in[0];
  const int*   ei = (const int*)d_in[1];
  const float* W1 = (const float*)d_in[2];
  const float* b1 = (const float*)d_in[3];
  const float* W2 = (const float*)d_in[4];
  const float* b2 = (const float*)d_in[5];

  const int n = in_sizes[0] / 64;   // 100000
  const int E = in_sizes[1] / 2;    // 1600000
  const int* src = ei;
  const int* dst = ei + E;

  char* ws = (char*)d_ws;
  size_t off = 0;
  auto carve = [&](size_t bytes) -> void* {
    off = (off + 255) & ~(size_t)255;
    void* p = ws + off;
    off += bytes;
    return p;
  };
  unsigned* deg  = (unsigned*)carve((size_t)n * 4);
  float*    dinv = (float*)carve((size_t)n * 4);
  float*    hs1  = (float*)carve((size_t)n * 64 * 4);  // reused as h1 after scatter1
  float*    agg1 = (float*)carve((size_t)n * 64 * 4);
  float*    hs2  = (float*)carve((size_t)n * 16 * 4);
  float*    agg2 = (float*)carve((size_t)n * 16 * 4);

  hipMemsetAsync(deg, 0, (size_t)n * 4, stream);
  k_deg<<<(E + 255) / 256, 256, 0, stream>>>(dst, deg, E);
  k_dinv<<<(n + 255) / 256, 256, 0, stream>>>(deg, dinv, n);

  const int tiles   = (n + 15) / 16;
  const int gblocks = (tiles + 7) / 8;

  k_gemm1<<<gblocks, 256, 0, stream>>>(x, W1, dinv, hs1, agg1, n);

  {
    long long tt = (long long)E * 16;
    k_scatter<<<(int)((tt + 255) / 256), 256, 0, stream>>>(src, dst, hs1, agg1, E, 4);
  }

  k_bias_relu<<<(n * 64 + 255) / 256, 256, 0, stream>>>(agg1, dinv, b1, hs1, n);

  k_gemm2<<<gblocks, 256, 0, stream>>>(hs1, W2, dinv, hs2, agg2, n);

  {
    long long tt = (long long)E * 4;
    k_scatter<<<(int)((tt + 255) / 256), 256, 0, stream>>>(src, dst, hs2, agg2, E, 2);
  }

  k_out<<<(n + 255) / 256, 256, 0, stream>>>(agg2, dinv, b2, (float*)d_out, n);
}